// PNA_88802743812678
// MI455X (gfx1250) — compile-verified
//
#include <hip/hip_runtime.h>

// CDNA5 / gfx1250. wave32. WMMA f32 16x16x4 (full-precision matrix path).

typedef __attribute__((ext_vector_type(2))) float v2f;
typedef __attribute__((ext_vector_type(8))) float v8f;

constexpr int DD    = 96;         // feature dim
constexpr int NTILE = DD / 16;    // 6 col tiles of 16
constexpr int PSTRIDE = 2 * DD + 32;   // 224 dwords per K-pair row (bank-swizzle:
                                       // 224 % 64 == 32 -> half-waves use disjoint banks)
constexpr int NPAIR = DD / 2;          // 48 K-pairs per 96-row weight panel
constexpr int SMWN  = NPAIR * PSTRIDE; // 10752 floats = 42 KB

__device__ __forceinline__ v8f wmma4(v2f a, v2f b, v8f c) {
    // D = A(16x4 f32) * B(4x16 f32) + C(16x16 f32)
    return __builtin_amdgcn_wmma_f32_16x16x4_f32(false, a, false, b, (short)0, c,
                                                 false, false);
}

// Cooperative: global row-major W[96][96] -> LDS K-pair-interleaved layout
//   smW[(k>>1)*PSTRIDE + col*2 + (k&1)]
__device__ __forceinline__ void load_weights_interleaved(
    float* __restrict__ smW, const float* __restrict__ W, int tid)
{
    for (int idx4 = tid; idx4 < DD * DD / 4; idx4 += 128) {
        const int k  = idx4 / (DD / 4);
        const int c0 = (idx4 % (DD / 4)) * 4;
        const float4 w = *(const float4*)(W + (size_t)k * DD + c0);
        float* s = smW + (k >> 1) * PSTRIDE + c0 * 2 + (k & 1);
        s[0] = w.x; s[2] = w.y; s[4] = w.z; s[6] = w.w;
    }
}

// ---------------------------------------------------------------------------
// Stage 1: h_self = x@W1 + b1 ; h_nb = x@W0 + b0   (two passes, one LDS panel)
// Block = 128 threads (4 waves), each wave owns a 16-row node tile.
// ---------------------------------------------------------------------------
__global__ __launch_bounds__(128) void gemm_dual(
    const float* __restrict__ x,
    const float* __restrict__ W1, const float* __restrict__ b1,
    const float* __restrict__ W0, const float* __restrict__ b0,
    float* __restrict__ hs, float* __restrict__ hn, int n)
{
    __shared__ float smW[SMWN];   // 42 KB
    __shared__ float smB[DD];

    const int tid  = threadIdx.x;
    const int lane = tid & 31;
    const int wv   = tid >> 5;
    const int half = lane >> 4;
    const int l16  = lane & 15;
    const int rowBase = blockIdx.x * 64 + wv * 16;

    int rowA = rowBase + l16;        // A-fragment row (M = lane%16)
    if (rowA >= n) rowA = n - 1;     // clamp; stores are guarded
    const float* xrow = x + (size_t)rowA * DD;

    for (int pass = 0; pass < 2; ++pass) {
        const float* W = pass ? W0 : W1;
        const float* B = pass ? b0 : b1;
        float* out     = pass ? hn : hs;

        __syncthreads();             // protect smW from previous pass readers
        load_weights_interleaved(smW, W, tid);
        if (tid < DD) smB[tid] = B[tid];
        __syncthreads();

        v8f acc[NTILE];
        #pragma unroll
        for (int t = 0; t < NTILE; ++t) {
            float bv = smB[t * 16 + l16];          // bias per output column
            acc[t] = (v8f){bv, bv, bv, bv, bv, bv, bv, bv};
        }

        for (int kk = 0; kk < DD; kk += 4) {
            // A 16x4 f32: lane holds K = kk + 2*half + {0,1}  (one b64 load)
            const float2 av = *(const float2*)(xrow + kk + 2 * half);
            v2f a; a.x = av.x; a.y = av.y;
            const float* bp = smW + ((kk >> 1) + half) * PSTRIDE + l16 * 2;
            #pragma unroll
            for (int t = 0; t < NTILE; ++t) {
                // B pair (K, K+1) contiguous in LDS -> single ds_load_b64
                const v2f b = *(const v2f*)(bp + t * 32);
                acc[t] = wmma4(a, b, acc[t]);
            }
        }

        #pragma unroll
        for (int t = 0; t < NTILE; ++t) {
            #pragma unroll
            for (int j = 0; j < 8; ++j) {
                int row = rowBase + half * 8 + j;  // D layout: M = 8*half + j
                if (row < n) out[(size_t)row * DD + t * 16 + l16] = acc[t][j];
            }
        }
    }
}

// ---------------------------------------------------------------------------
// Zero the aggregation buffer (float4 grid-stride).
// ---------------------------------------------------------------------------
__global__ void zero_f4(float4* __restrict__ p, size_t n4)
{
    size_t i = (size_t)blockIdx.x * blockDim.x + threadIdx.x;
    size_t stride = (size_t)gridDim.x * blockDim.x;
    float4 z = {0.f, 0.f, 0.f, 0.f};
    for (; i < n4; i += stride) p[i] = z;
}

// ---------------------------------------------------------------------------
// Edge scatter: agg[dst] += h_nb[src].  edge_dst is sorted -> run-length
// accumulate in registers, flush with one atomicAdd per run boundary
// (avg run 800K/50K = 16 -> ~16x fewer f32 atomics).
// 96 threads: thread t owns column t (coalesced 384B row reads, L2-resident).
// ---------------------------------------------------------------------------
__global__ __launch_bounds__(96) void scatter_edges(
    const float* __restrict__ hn,
    const int* __restrict__ esrc, const int* __restrict__ edst,
    float* __restrict__ agg, int nEdges)
{
    const int col = threadIdx.x;
    const int e0  = blockIdx.x * 512;
    int e1 = e0 + 512; if (e1 > nEdges) e1 = nEdges;
    if (e0 >= e1) return;

    int prev = edst[e0];           // uniform across block -> scalar loads
    float accv = 0.f;
    for (int e = e0; e < e1; ++e) {
        const int d = edst[e];
        const int s = esrc[e];
        if (e + 16 < e1)
            __builtin_prefetch(hn + (size_t)esrc[e + 16] * DD + col, 0, 3);
        const float v = hn[(size_t)s * DD + col];
        if (d != prev) {           // uniform branch
            atomicAdd(&agg[(size_t)prev * DD + col], accv);
            prev = d;
            accv = v;
        } else {
            accv += v;
        }
    }
    atomicAdd(&agg[(size_t)prev * DD + col], accv);
}

// ---------------------------------------------------------------------------
// Stage 2: y = [h_self | agg] @ W2 + b2 ; out = y / std(y, ddof=1)
// K=192 as two K-halves reusing one LDS panel; accumulators persist.
// Row std computed in-registers via shfl_xor butterfly (width 16).
// ---------------------------------------------------------------------------
__global__ __launch_bounds__(128) void gemm_cat_norm(
    const float* __restrict__ hs, const float* __restrict__ agg,
    const float* __restrict__ W2, const float* __restrict__ b2,
    float* __restrict__ out, int n)
{
    __shared__ float smW[SMWN];   // one K-half of W2: 96x96 panel

    const int tid  = threadIdx.x;
    const int lane = tid & 31;
    const int wv   = tid >> 5;
    const int half = lane >> 4;
    const int l16  = lane & 15;
    const int rowBase = blockIdx.x * 64 + wv * 16;

    int rowA = rowBase + l16;
    if (rowA >= n) rowA = n - 1;

    v8f acc[NTILE];
    #pragma unroll
    for (int t = 0; t < NTILE; ++t) {
        float bv = b2[t * 16 + l16];
        acc[t] = (v8f){bv, bv, bv, bv, bv, bv, bv, bv};
    }

    for (int kh = 0; kh < 2; ++kh) {
        __syncthreads();
        load_weights_interleaved(smW, W2 + (size_t)kh * DD * DD, tid);
        __syncthreads();

        const float* xrow = (kh ? agg : hs) + (size_t)rowA * DD;
        for (int kk = 0; kk < DD; kk += 4) {
            const float2 av = *(const float2*)(xrow + kk + 2 * half);
            v2f a; a.x = av.x; a.y = av.y;
            const float* bp = smW + ((kk >> 1) + half) * PSTRIDE + l16 * 2;
            #pragma unroll
            for (int t = 0; t < NTILE; ++t) {
                const v2f b = *(const v2f*)(bp + t * 32);
                acc[t] = wmma4(a, b, acc[t]);
            }
        }
    }

    // Row r = rowBase + 8*half + j lives across lanes (lane%16, same half),
    // components acc[t][j], t = 0..5.
    #pragma unroll
    for (int j = 0; j < 8; ++j) {
        float s1 = 0.f, s2 = 0.f;
        #pragma unroll
        for (int t = 0; t < NTILE; ++t) {
            const float v = acc[t][j];
            s1 += v; s2 += v * v;
        }
        #pragma unroll
        for (int m = 1; m < 16; m <<= 1) {
            s1 += __shfl_xor(s1, m, 16);
            s2 += __shfl_xor(s2, m, 16);
        }
        const float mean = s1 * (1.0f / 96.0f);
        const float var  = (s2 - 96.0f * mean * mean) * (1.0f / 95.0f);
        const float inv  = rsqrtf(var);
        #pragma unroll
        for (int t = 0; t < NTILE; ++t) acc[t][j] *= inv;
    }

    #pragma unroll
    for (int t = 0; t < NTILE; ++t) {
        #pragma unroll
        for (int j = 0; j < 8; ++j) {
            const int row = rowBase + half * 8 + j;
            if (row < n) out[(size_t)row * DD + t * 16 + l16] = acc[t][j];
        }
    }
}

// ---------------------------------------------------------------------------
extern "C" void kernel_launch(void* const* d_in, const int* in_sizes, int n_in,
                              void* d_out, int out_size, void* d_ws, size_t ws_size,
                              hipStream_t stream)
{
    (void)n_in; (void)out_size;
    const float* x   = (const float*)d_in[0];
    const float* W0  = (const float*)d_in[1];
    const float* b0  = (const float*)d_in[2];
    const float* W1  = (const float*)d_in[3];
    const float* b1  = (const float*)d_in[4];
    const float* W2  = (const float*)d_in[5];
    const float* b2  = (const float*)d_in[6];
    const int*   esrc = (const int*)d_in[7];
    const int*   edst = (const int*)d_in[8];

    const int n     = in_sizes[0] / DD;
    const int E     = in_sizes[7];
    const int depth = in_sizes[1] / (DD * DD);
    const size_t nd = (size_t)n * DD;

    float* hs  = (float*)d_ws;
    float* hn  = hs + nd;
    float* agg = hn + nd;
    // 4th buffer if workspace allows, else alias hn (safe: each wave reads only
    // its own 16 rows before writing them, and hn is dead after scatter).
    float* xb  = (ws_size >= 4 * nd * sizeof(float)) ? (agg + nd) : hn;

    const dim3 blk(128);
    const dim3 grd((n + 63) / 64);

    const size_t n4 = nd / 4;
    int zblocks = (int)((n4 + 255) / 256);
    if (zblocks > 4096) zblocks = 4096;

    for (int i = 0; i < depth; ++i) {
        const float* xin = (i == 0) ? x : xb;
        float* xout = (i == depth - 1) ? (float*)d_out : xb;

        gemm_dual<<<grd, blk, 0, stream>>>(
            xin, W1 + (size_t)i * DD * DD, b1 + (size_t)i * DD,
                 W0 + (size_t)i * DD * DD, b0 + (size_t)i * DD, hs, hn, n);

        zero_f4<<<zblocks, 256, 0, stream>>>((float4*)agg, n4);

        scatter_edges<<<(E + 511) / 512, 96, 0, stream>>>(hn, esrc, edst, agg, E);

        gemm_cat_norm<<<grd, blk, 0, stream>>>(
            hs, agg, W2 + (size_t)i * 2 * DD * DD, b2 + (size_t)i * DD, xout, n);
    }
}